// GSSA_kan_56152402428566
// MI455X (gfx1250) — compile-verified
//
#include <hip/hip_runtime.h>
#include <math.h>

// ---------------------------------------------------------------------------
// GSSA + KAN pipeline for MI455X (gfx1250), FP32 via V_WMMA_F32_16X16X4_F32.
// KAN pointwise layers are dense GEMMs over the expanded feature basis
// phi = [silu(x); rbf_0..7(x)], K = 9*96 = 864, with weights pre-packed into
// a contiguous Wcat[Cout][864] so A-fragments are single b64 loads.
// ---------------------------------------------------------------------------

typedef float v2f __attribute__((ext_vector_type(2)));
typedef float v8f __attribute__((ext_vector_type(8)));

#define DIM    96
#define HEADS  3
#define DHEAD  32
#define NTOK   17                  // g*g + 1
#define NW     64                  // windows per image (8x8)
#define BATCH  128
#define BP     (BATCH*NW)          // 8192 windows
#define L1     (BP*NTOK)           // 139264 qkv positions
#define L2     BP                  // 8192 group-token positions
#define L3     (BATCH*1024)        // 131072 output-map positions
#define KDIM   (9*DIM)             // 864 expanded features
#define ATS    (NW+2)              // padded attn row stride (bank-conflict free)
#define SCALE  0.1767766952966369f // 32^-0.5
#define RBF_H    (4.0f/7.0f)
#define RBF_INVH (7.0f/4.0f)

__device__ __forceinline__ float siluf(float v) { return v / (1.0f + __expf(-v)); }

// ---------------------------------------------------------------------------
// Pack [wb | ws] into contiguous Wcat[Cout][864], k = f*96 + ch (f=0: silu wt).
// ---------------------------------------------------------------------------
__global__ __launch_bounds__(256) void pack_wcat(const float* __restrict__ wb,
                                                 const float* __restrict__ ws,
                                                 float* __restrict__ wcat, int cout)
{
    int idx = blockIdx.x * 256 + threadIdx.x;
    if (idx >= cout * KDIM) return;
    int row = idx / KDIM, k = idx - row * KDIM;
    int f = k / DIM, i = k - f * DIM;
    wcat[idx] = (f == 0) ? wb[row * DIM + i] : ws[(row * DIM + i) * 8 + (f - 1)];
}

// ---------------------------------------------------------------------------
// Build windowed input, channel-major: xw[ch][l], l = window*17 + slot.
// ---------------------------------------------------------------------------
__global__ __launch_bounds__(256) void build_xw(const float* __restrict__ x,
                                                const float* __restrict__ gtok,
                                                float* __restrict__ xw)
{
    int ch = blockIdx.y;
    int l  = blockIdx.x * 256 + threadIdx.x;
    if (l >= L1) return;
    int w = l / NTOK, s = l - w * NTOK;
    float v;
    if (s == 0) {
        v = gtok[ch];
    } else {
        int p  = s - 1;
        int b  = w >> 6,  win = w & 63;
        int wx = win >> 3, wy = win & 7;
        int px = p >> 2,   py = p & 3;
        int hh = wx * 4 + px, ww = wy * 4 + py;
        v = x[(((size_t)b * DIM + ch) * 32 + hh) * 32 + ww];
    }
    xw[(size_t)ch * L1 + l] = v;
}

// ---------------------------------------------------------------------------
// KAN GEMM: out(Cout x L) = Wcat(Cout x 864) @ phi(864 x L).
// One workgroup per 16-column tile; phi staged in LDS (55 KB).
// Exactly COUT/MPW waves per block; each wave owns MPW rows (MPW/16 C-tiles),
// reusing the B-fragment across the row-subtiles.
// Output index: out[(pos/SPATIAL)*Cout*SPATIAL + row*SPATIAL + pos%SPATIAL].
// ---------------------------------------------------------------------------
template<int COUT, int SPATIAL, int MPW>
__global__ __launch_bounds__(32 * (COUT / MPW))
void kan_gemm(const float* __restrict__ xin, const float* __restrict__ wcat,
              float* __restrict__ out, int L)
{
    constexpr int NWAVE = COUT / MPW;
    constexpr int TPB   = 32 * NWAVE;
    __shared__ float phi[KDIM * 16];          // 55296 bytes
    const int tid  = threadIdx.x;
    const int lane = tid & 31;
    const int wv   = tid >> 5;                // wave index: owns rows wv*MPW..+MPW-1 (wave-uniform)
    const int base = blockIdx.x * 16;

    // Stage expanded features for this 16-position tile.
    for (int p = tid; p < DIM * 16; p += TPB) {
        int ch = p >> 4, col = p & 15;
        int pos = base + col;
        float v = (pos < L) ? xin[(size_t)ch * L + pos] : 0.0f;
        phi[ch * 16 + col] = siluf(v);
        #pragma unroll
        for (int f = 0; f < 8; ++f) {
            float c = -2.0f + f * RBF_H;
            float t = (v - c) * RBF_INVH;
            phi[(DIM + f * DIM + ch) * 16 + col] = __expf(-t * t);
        }
    }
    __syncthreads();

    // V_WMMA_F32_16X16X4_F32 fragment layout (ISA 7.12.2):
    // lanes 0-15: K = {k0, k0+1}; lanes 16-31: K = {k0+2, k0+3}.
    const int m    = lane & 15;               // A row / B column within tile
    const int krel = (lane >> 4) << 1;        // 0 or 2
    const int rofs = (lane >> 4) << 3;        // C: M = r or r+8

    v8f acc[MPW / 16] = {};
    const float* wrow = wcat + (size_t)(wv * MPW + m) * KDIM;

    for (int k0 = 0; k0 < KDIM; k0 += 4) {
        int ka = k0 + krel;
        v2f bf;
        bf.x = phi[ka * 16 + m];
        bf.y = phi[(ka + 1) * 16 + m];
        #pragma unroll
        for (int s = 0; s < MPW / 16; ++s) {
            v2f a = *(const v2f*)(wrow + (size_t)s * 16 * KDIM + ka);  // b64 load
            acc[s] = __builtin_amdgcn_wmma_f32_16x16x4_f32(
                false, a, false, bf, (short)0, acc[s], false, false);
        }
    }

    int pos = base + m;
    if (pos < L) {
        int bo = pos / SPATIAL, sp = pos - bo * SPATIAL;
        #pragma unroll
        for (int s = 0; s < MPW / 16; ++s)
            #pragma unroll
            for (int r = 0; r < 8; ++r) {
                int mm = wv * MPW + s * 16 + r + rofs;
                out[((size_t)bo * COUT + mm) * SPATIAL + sp] = acc[s][r];
            }
    }
}

// ---------------------------------------------------------------------------
// 17-token softmax attention per (window, head). One wave per block.
// gt:  [b,h][w][32]   group tokens
// gfT: [b,h][n=512][j=64]  grid features TRANSPOSED (n = p*32+d, j = window)
// so the wattn GEMM can load B-fragment (k,k+1) pairs as one b64.
// ---------------------------------------------------------------------------
__global__ __launch_bounds__(32) void attn17(const float* __restrict__ qkv,
                                             float* __restrict__ gt,
                                             float* __restrict__ gfT)
{
    __shared__ float qs[NTOK * 32], ks_[NTOK * 32], vs[NTOK * 32];
    int wi = blockIdx.x / HEADS, h = blockIdx.x % HEADS;
    int tid = threadIdx.x;
    for (int i = 0; i < NTOK; ++i) {
        size_t l = (size_t)wi * NTOK + i;
        qs[i * 32 + tid]  = qkv[(size_t)(h * 32 + tid) * L1 + l] * SCALE;
        ks_[i * 32 + tid] = qkv[(size_t)(DIM + h * 32 + tid) * L1 + l];
        vs[i * 32 + tid]  = qkv[(size_t)(2 * DIM + h * 32 + tid) * L1 + l];
    }
    __syncthreads();
    if (tid < NTOK) {
        int i = tid;
        float d[NTOK], mx = -1e30f;
        for (int j = 0; j < NTOK; ++j) {
            float s = 0.0f;
            for (int dd = 0; dd < 32; ++dd) s += qs[i * 32 + dd] * ks_[j * 32 + dd];
            d[j] = s; mx = fmaxf(mx, s);
        }
        float sum = 0.0f;
        for (int j = 0; j < NTOK; ++j) { d[j] = __expf(d[j] - mx); sum += d[j]; }
        float inv = 1.0f / sum;
        int b = wi >> 6, w = wi & 63;
        size_t bh = (size_t)(b * HEADS + h);
        for (int dd = 0; dd < 32; ++dd) {
            float o = 0.0f;
            for (int j = 0; j < NTOK; ++j) o += d[j] * vs[j * 32 + dd];
            o *= inv;
            if (i == 0)
                gt[(bh * NW + w) * 32 + dd] = o;
            else
                gfT[(bh * 512 + (size_t)(i - 1) * 32 + dd) * NW + w] = o;
        }
    }
}

// ---------------------------------------------------------------------------
// LayerNorm(32) + exact GELU on group tokens; wave32 butterfly reductions.
// ---------------------------------------------------------------------------
__global__ __launch_bounds__(32) void ln_gelu(const float* __restrict__ gt,
                                              const float* __restrict__ lng,
                                              const float* __restrict__ lnb,
                                              float* __restrict__ hh)
{
    int g = blockIdx.x, d = threadIdx.x;
    float v = gt[(size_t)g * 32 + d];
    float s = v;
    for (int o = 16; o; o >>= 1) s += __shfl_xor(s, o, 32);
    float mu = s * (1.0f / 32.0f);
    float df = v - mu;
    float s2 = df * df;
    for (int o = 16; o; o >>= 1) s2 += __shfl_xor(s2, o, 32);
    float var = s2 * (1.0f / 32.0f);
    float hn = df * rsqrtf(var + 1e-5f) * lng[d] + lnb[d];
    float ge = 0.5f * hn * (1.0f + erff(hn * 0.70710678118654752f));
    int b = g / (HEADS * NW), r = g - b * HEADS * NW;
    int h = r / NW, w = r - h * NW;
    hh[(size_t)(h * 32 + d) * L2 + b * NW + w] = ge;
}

// ---------------------------------------------------------------------------
// Window attention per (b,h): 64x64 softmax (VALU) then WMMA GEMM
// attn(64x64) @ gf(64x512) with A from padded LDS (b64, conflict-free) and
// B from transposed gfT (b64 global). Scatter into channel-major fmap.
// ---------------------------------------------------------------------------
__global__ __launch_bounds__(256) void wattn(const float* __restrict__ qk,
                                             const float* __restrict__ gfT,
                                             float* __restrict__ fmap)
{
    __shared__ float Qs[NW * 32], Ks[NW * 32], At[NW * ATS];   // ~32.5 KB
    int b = blockIdx.x / HEADS, h = blockIdx.x % HEADS;
    int tid = threadIdx.x, lane = tid & 31, wv = tid >> 5;

    for (int idx = tid; idx < NW * 32; idx += 256) {
        int j = idx >> 5, d = idx & 31;
        Qs[idx] = qk[(size_t)(h * 64 + d) * L2 + b * NW + j] * SCALE;
        Ks[idx] = qk[(size_t)(h * 64 + 32 + d) * L2 + b * NW + j];
    }
    __syncthreads();

    if (tid < NW) {
        int i = tid; float mx = -1e30f;
        for (int j = 0; j < NW; ++j) {
            float s = 0.0f;
            for (int dd = 0; dd < 32; ++dd) s += Qs[i * 32 + dd] * Ks[j * 32 + dd];
            At[i * ATS + j] = s; mx = fmaxf(mx, s);
        }
        float sum = 0.0f;
        for (int j = 0; j < NW; ++j) { float e = __expf(At[i * ATS + j] - mx); At[i * ATS + j] = e; sum += e; }
        float inv = 1.0f / sum;
        for (int j = 0; j < NW; ++j) At[i * ATS + j] *= inv;
    }
    __syncthreads();

    const float* gfp = gfT + (size_t)(b * HEADS + h) * 512 * NW;
    const int m    = lane & 15;
    const int krel = (lane >> 4) << 1;
    const int rofs = (lane >> 4) << 3;

    for (int t = wv; t < 128; t += 8) {           // 4 M-tiles x 32 N-tiles
        int mt = t >> 5, nt = t & 31;
        int ncol = nt * 16 + m;
        v8f acc = {};
        for (int k0 = 0; k0 < NW; k0 += 4) {
            int ka = k0 + krel;
            v2f a  = *(const v2f*)(&At[(mt * 16 + m) * ATS + ka]);       // LDS b64
            v2f bb = *(const v2f*)(gfp + (size_t)ncol * NW + ka);        // global b64
            acc = __builtin_amdgcn_wmma_f32_16x16x4_f32(
                false, a, false, bb, (short)0, acc, false, false);
        }
        #pragma unroll
        for (int r = 0; r < 8; ++r) {
            int i = mt * 16 + r + rofs;           // output window index
            int p = ncol >> 5, d = ncol & 31;     // (pixel, dhead)
            int wx = i >> 3, wy = i & 7, px = p >> 2, py = p & 3;
            int hh_ = wx * 4 + px, ww_ = wy * 4 + py;
            fmap[(size_t)(h * 32 + d) * L3 + b * 1024 + hh_ * 32 + ww_] = acc[r];
        }
    }
}

// ---------------------------------------------------------------------------
extern "C" void kernel_launch(void* const* d_in, const int* in_sizes, int n_in,
                              void* d_out, int out_size, void* d_ws, size_t ws_size,
                              hipStream_t stream)
{
    const float* x      = (const float*)d_in[0];
    const float* gtok   = (const float*)d_in[1];
    const float* qkv_wb = (const float*)d_in[2];
    const float* qkv_ws = (const float*)d_in[3];
    const float* ln_g   = (const float*)d_in[4];
    const float* ln_b   = (const float*)d_in[5];
    const float* gt_wb  = (const float*)d_in[6];
    const float* gt_ws  = (const float*)d_in[7];
    const float* out_wb = (const float*)d_in[8];
    const float* out_ws = (const float*)d_in[9];
    float* out = (float*)d_out;

    // Workspace layout (floats), ~329 MB total.
    float* wsf   = (float*)d_ws;
    float* xw    = wsf;                                      // 96  * L1
    float* qkv   = xw    + (size_t)DIM * L1;                 // 288 * L1
    float* gt    = qkv   + (size_t)(3 * DIM) * L1;           // 3*128*64*32
    float* gfT   = gt    + (size_t)HEADS * BATCH * NW * 32;  // 3*128*512*64
    float* hh    = gfT   + (size_t)HEADS * BATCH * 512 * NW; // 96  * L2
    float* qk    = hh    + (size_t)DIM * L2;                 // 192 * L2
    float* fmap  = qk    + (size_t)(2 * DIM) * L2;           // 96  * L3
    float* wcatQ = fmap  + (size_t)DIM * L3;                 // 288 * 864
    float* wcatG = wcatQ + (size_t)(3 * DIM) * KDIM;         // 192 * 864
    float* wcatO = wcatG + (size_t)(2 * DIM) * KDIM;         // 96  * 864

    pack_wcat<<<(3 * DIM * KDIM + 255) / 256, 256, 0, stream>>>(qkv_wb, qkv_ws, wcatQ, 3 * DIM);
    pack_wcat<<<(2 * DIM * KDIM + 255) / 256, 256, 0, stream>>>(gt_wb, gt_ws, wcatG, 2 * DIM);
    pack_wcat<<<(DIM * KDIM + 255) / 256, 256, 0, stream>>>(out_wb, out_ws, wcatO, DIM);

    build_xw<<<dim3((L1 + 255) / 256, DIM), 256, 0, stream>>>(x, gtok, xw);

    kan_gemm<3 * DIM, L1, 32><<<(L1 + 15) / 16, 288, 0, stream>>>(xw, wcatQ, qkv, L1);
    attn17<<<BP * HEADS, 32, 0, stream>>>(qkv, gt, gfT);
    ln_gelu<<<BATCH * HEADS * NW, 32, 0, stream>>>(gt, ln_g, ln_b, hh);
    kan_gemm<2 * DIM, L2, 32><<<(L2 + 15) / 16, 192, 0, stream>>>(hh, wcatG, qk, L2);
    wattn<<<BATCH * HEADS, 256, 0, stream>>>(qk, gfT, fmap);
    kan_gemm<DIM, 1024, 16><<<(L3 + 15) / 16, 192, 0, stream>>>(fmap, wcatO, out, L3);
}